// RelGraphConvOps_27908697489554
// MI455X (gfx1250) — compile-verified
//
#include <hip/hip_runtime.h>

typedef __attribute__((ext_vector_type(2))) float v2f;
typedef __attribute__((ext_vector_type(8))) float v8f;

#define IN_FEAT   64
#define OUT_FEAT  64
#define NUM_BASES 2

// ---------------------------------------------------------------------------
// Fused GEMM: out-cols tiled over [W0 | W1 | loop_weight] (3 x 64x64, 12 col
// tiles of 16). sel 0/1 -> P (d_ws, node-major 128 wide), sel 2 -> d_out with
// bias added. Uses native f32 WMMA 16x16x4, K=64 => 16 chained WMMAs / tile.
// ---------------------------------------------------------------------------
__global__ __launch_bounds__(256) void rgcn_gemm_kernel(
    const float* __restrict__ feat, const float* __restrict__ W,
    const float* __restrict__ loop_w, const float* __restrict__ bias,
    float* __restrict__ P, float* __restrict__ hout, int row_tiles)
{
  const int lane = threadIdx.x & 31;
  const int wave = threadIdx.x >> 5;          // 8 waves / block
  const int ct   = blockIdx.x;                // 0..11 column tile
  const int sel  = ct >> 2;                   // 0:W0  1:W1  2:loop_weight
  const int lc   = ct & 3;                    // local 16-col tile in B
  const float* __restrict__ B =
      (sel == 0) ? W : (sel == 1) ? (W + IN_FEAT * OUT_FEAT) : loop_w;

  const int half = lane >> 4;                 // 0: lanes 0-15, 1: lanes 16-31
  const int l15  = lane & 15;
  const int colb = lc * 16 + l15;             // B/D column owned by this lane

  // Preload all 16 B k-fragments (ISA layout: N=lane&15, K=k0+half*2+{0,1})
  v2f bfrag[16];
#pragma unroll
  for (int s = 0; s < 16; ++s) {
    const int kb = s * 4 + half * 2;
    bfrag[s].x = B[kb * OUT_FEAT + colb];
    bfrag[s].y = B[(kb + 1) * OUT_FEAT + colb];
  }
  const float bcol = bias[colb & 63];

  for (int rt = blockIdx.y * 8 + wave; rt < row_tiles; rt += gridDim.y * 8) {
    const int rowA = rt * 16 + l15;           // A row this lane supplies
    v8f acc = {0.f, 0.f, 0.f, 0.f, 0.f, 0.f, 0.f, 0.f};
#pragma unroll
    for (int s = 0; s < 16; ++s) {
      // A layout: M=lane&15, K=s*4 + half*2 + {0,1} -> contiguous float2
      const v2f a = *(const v2f*)(feat + (size_t)rowA * IN_FEAT + s * 4 + half * 2);
      acc = __builtin_amdgcn_wmma_f32_16x16x4_f32(false, a, false, bfrag[s],
                                                  (short)0, acc, false, false);
    }
    // D layout: M = rt*16 + half*8 + i, N = colb
    const int mrow = rt * 16 + half * 8;
    if (sel == 2) {
#pragma unroll
      for (int i = 0; i < 8; ++i)
        hout[(size_t)(mrow + i) * OUT_FEAT + colb] = acc[i] + bcol;
    } else {
      float* dst = P + (size_t)mrow * (2 * OUT_FEAT) + sel * OUT_FEAT + colb;
#pragma unroll
      for (int i = 0; i < 8; ++i)
        dst[(size_t)i * (2 * OUT_FEAT)] = acc[i];
    }
  }
}

// ---------------------------------------------------------------------------
// Edge stage: h[dst] += c0*P0[src] + c1*P1[src].  64 lanes per edge, one
// feature each; f32 atomic add (no return) into L2-resident h.
// ---------------------------------------------------------------------------
__global__ __launch_bounds__(256) void rgcn_edge_scatter_kernel(
    const float* __restrict__ P, const float* __restrict__ coeff,
    const int* __restrict__ src, const int* __restrict__ dst,
    const int* __restrict__ ety, float* __restrict__ h, int n_edges)
{
  const int e = blockIdx.x * 4 + (threadIdx.x >> 6);
  if (e >= n_edges) return;
  const int f = threadIdx.x & 63;
  const int s = src[e];
  const int d = dst[e];
  const int r = ety[e];
  const float c0 = coeff[r * NUM_BASES + 0];
  const float c1 = coeff[r * NUM_BASES + 1];
  const float* __restrict__ p = P + (size_t)s * (2 * OUT_FEAT);
  const float v = c0 * p[f] + c1 * p[OUT_FEAT + f];
  atomicAdd(h + (size_t)d * OUT_FEAT + f, v);
}

__global__ __launch_bounds__(256) void rgcn_relu_kernel(float* __restrict__ h, int n)
{
  const int i = blockIdx.x * blockDim.x + threadIdx.x;
  if (i < n) {
    const float v = h[i];
    h[i] = v > 0.f ? v : 0.f;
  }
}

// ---------------------------------------------------------------------------
// Inputs (setup_inputs order): feat, coeff, W, h_bias, loop_weight, src_ids,
// dst_ids, etypes.  d_ws holds P (n_nodes x 128 f32 = 51.2 MB).
// ---------------------------------------------------------------------------
extern "C" void kernel_launch(void* const* d_in, const int* in_sizes, int n_in,
                              void* d_out, int out_size, void* d_ws, size_t ws_size,
                              hipStream_t stream)
{
  const float* feat   = (const float*)d_in[0];
  const float* coeff  = (const float*)d_in[1];
  const float* W      = (const float*)d_in[2];
  const float* bias   = (const float*)d_in[3];
  const float* loop_w = (const float*)d_in[4];
  const int*   src    = (const int*)d_in[5];
  const int*   dst    = (const int*)d_in[6];
  const int*   ety    = (const int*)d_in[7];
  float* h = (float*)d_out;
  float* P = (float*)d_ws;

  const int n_nodes  = in_sizes[0] / IN_FEAT;   // 100000 (16 | n_nodes)
  const int n_edges  = in_sizes[5];             // 1600000
  const int row_tiles = n_nodes / 16;           // 6250

  // 1) P0,P1 = feat@W0, feat@W1 (to d_ws); h = bias + feat@loop_w (to d_out)
  dim3 g1(12, 104);   // 12 col tiles x (104*8 wave slots striding row tiles)
  rgcn_gemm_kernel<<<g1, 256, 0, stream>>>(feat, W, loop_w, bias, P, h, row_tiles);

  // 2) per-edge scatter-add into h
  const int eblocks = (n_edges + 3) / 4;
  rgcn_edge_scatter_kernel<<<eblocks, 256, 0, stream>>>(P, coeff, src, dst, ety, h, n_edges);

  // 3) relu
  const int n = out_size;
  rgcn_relu_kernel<<<(n + 255) / 256, 256, 0, stream>>>(h, n);
}